// MI_estimator_75634374083349
// MI455X (gfx1250) — compile-verified
//
#include <hip/hip_runtime.h>
#include <hip/hip_bf16.h>
#include <math.h>

// Problem constants (from reference)
#define NN 8192
#define DD 256
#define NG 512
#define EPSV 1e-6f

typedef __attribute__((ext_vector_type(2))) float v2f;
typedef __attribute__((ext_vector_type(8))) float v8f;
typedef __attribute__((ext_vector_type(4))) unsigned int v4u;
typedef __attribute__((ext_vector_type(8))) int v8i;
typedef __attribute__((ext_vector_type(4))) int v4i;

#define LDSS 260   // padded LDS row stride (floats); 260 % 64 == 4 -> conflict-free b64 frags

// ---------------------------------------------------------------------------
// Kernel 1: OUT = relu(X @ W1 + b1) @ W2 + b2 ; norms[i] = ||OUT[i,:]||
// One workgroup = 64 rows. 256 threads = 8 waves. X tile staged into LDS via
// the Tensor Data Mover (D# descriptor with LDS padding -> stride 260),
// hidden kept in LDS. All FLOPs via v_wmma_f32_16x16x4_f32.
// ---------------------------------------------------------------------------
__global__ __launch_bounds__(256) void mlp_norm_kernel(
    const float* __restrict__ X,
    const float* __restrict__ W1, const float* __restrict__ bias1,
    const float* __restrict__ W2, const float* __restrict__ bias2,
    float* __restrict__ OUT, float* __restrict__ norms)
{
    __shared__ float Xs[64 * LDSS];   // input tile, reused for output tile (norms)
    __shared__ float Hs[64 * LDSS];   // relu hidden, reused for norm partials

    const int tid  = threadIdx.x;
    const int wave = tid >> 5;
    const int lane = tid & 31;
    const int half = lane >> 4;
    const int l16  = lane & 15;
    const int row0 = blockIdx.x * 64;

#if __has_builtin(__builtin_amdgcn_tensor_load_to_lds)
    // --- TDM: DMA the 64x256 f32 tile into LDS with 4-dword pad per 256 dwords
    //     (=> padded row stride of 260 floats, matching LDSS). One wave issues;
    //     TENSORcnt tracks completion; barrier publishes to the workgroup.
    if (wave == 0) {
        unsigned long long gaddr = (unsigned long long)(const void*)(X + (size_t)row0 * DD);
        unsigned int ldsa = (unsigned int)(uintptr_t)(void*)(&Xs[0]); // LDS aperture: low 32 bits = LDS byte offset
        v4u g0;
        g0[0] = 1u;                                                   // count=1 (valid user D#)
        g0[1] = ldsa;                                                 // lds_addr [63:32]
        g0[2] = (unsigned int)gaddr;                                  // global_addr[31:0]
        g0[3] = (unsigned int)((gaddr >> 32) & 0x01FFFFFFu) | (2u << 30); // addr[56:32] | type=2
        v8i g1;
        g1[0] = (int)((2u << 16)      // data_size = 4 bytes
                    | (1u << 20)      // pad_enable
                    | (7u << 22)      // pad_interval: 256 dwords
                    | (3u << 25));    // pad_amount:   4 dwords
        g1[1] = (int)(256u << 16);    // tensor_dim0 = 256   ([63:48] = low bits)
        g1[2] = (int)(8192u << 16);   // tensor_dim1 = 8192  ([95:80] = low bits)
        g1[3] = (int)(256u << 16);    // tile_dim0 = 256     ([127:112])
        g1[4] = 64;                   // tile_dim1 = 64      ([143:128])
        g1[5] = 256;                  // tensor_dim0_stride = 256 elements
        g1[6] = 0;
        g1[7] = 0;
        v4i zz = {0, 0, 0, 0};
#if __clang_major__ >= 23
        v8i z8 = {0, 0, 0, 0, 0, 0, 0, 0};
        __builtin_amdgcn_tensor_load_to_lds(g0, g1, zz, zz, z8, 0);
#else
        __builtin_amdgcn_tensor_load_to_lds(g0, g1, zz, zz, 0);
#endif
        __builtin_amdgcn_s_wait_tensorcnt(0);
    }
#else
    // Fallback: manual float4 staging with explicit padded stride.
    for (int i = tid * 4; i < 64 * 256; i += 256 * 4) {
        int r = i >> 8;
        int c = i & 255;
        float4 v = *(const float4*)(X + (size_t)(row0 + r) * DD + c);
        *(float4*)(&Xs[r * LDSS + c]) = v;
    }
#endif
    __syncthreads();

    const int nt0 = wave * 2;          // each wave owns two 16-col output tiles
    const int nt1 = nt0 + 1;

    // ---------------- GEMM 1: H = relu(X @ W1 + b1) ----------------
    v8f acc[4][2] = {};
#pragma unroll 2
    for (int kk = 0; kk < 64; ++kk) {
        const int kl = kk * 4 + 2 * half;
        v2f a[4];
#pragma unroll
        for (int mt = 0; mt < 4; ++mt)
            a[mt] = *(const v2f*)&Xs[(mt * 16 + l16) * LDSS + kl];
        v2f b0, b1f;
        b0.x  = W1[(size_t)kl * DD + nt0 * 16 + l16];
        b0.y  = W1[(size_t)(kl + 1) * DD + nt0 * 16 + l16];
        b1f.x = W1[(size_t)kl * DD + nt1 * 16 + l16];
        b1f.y = W1[(size_t)(kl + 1) * DD + nt1 * 16 + l16];
#pragma unroll
        for (int mt = 0; mt < 4; ++mt) {
            acc[mt][0] = __builtin_amdgcn_wmma_f32_16x16x4_f32(
                false, a[mt], false, b0,  (short)0, acc[mt][0], false, false);
            acc[mt][1] = __builtin_amdgcn_wmma_f32_16x16x4_f32(
                false, a[mt], false, b1f, (short)0, acc[mt][1], false, false);
        }
    }
    {
        const float bb0 = bias1[nt0 * 16 + l16];
        const float bb1 = bias1[nt1 * 16 + l16];
#pragma unroll
        for (int mt = 0; mt < 4; ++mt) {
#pragma unroll
            for (int r = 0; r < 8; ++r) {
                int rr = mt * 16 + r + 8 * half;
                float h0 = acc[mt][0][r] + bb0; h0 = h0 > 0.0f ? h0 : 0.0f;
                float h1 = acc[mt][1][r] + bb1; h1 = h1 > 0.0f ? h1 : 0.0f;
                Hs[rr * LDSS + nt0 * 16 + l16] = h0;
                Hs[rr * LDSS + nt1 * 16 + l16] = h1;
            }
        }
    }
    __syncthreads();

    // ---------------- GEMM 2: OUT = H @ W2 + b2 ----------------
    v8f acc2[4][2] = {};
#pragma unroll 2
    for (int kk = 0; kk < 64; ++kk) {
        const int kl = kk * 4 + 2 * half;
        v2f a[4];
#pragma unroll
        for (int mt = 0; mt < 4; ++mt)
            a[mt] = *(const v2f*)&Hs[(mt * 16 + l16) * LDSS + kl];
        v2f b0, b1f;
        b0.x  = W2[(size_t)kl * DD + nt0 * 16 + l16];
        b0.y  = W2[(size_t)(kl + 1) * DD + nt0 * 16 + l16];
        b1f.x = W2[(size_t)kl * DD + nt1 * 16 + l16];
        b1f.y = W2[(size_t)(kl + 1) * DD + nt1 * 16 + l16];
#pragma unroll
        for (int mt = 0; mt < 4; ++mt) {
            acc2[mt][0] = __builtin_amdgcn_wmma_f32_16x16x4_f32(
                false, a[mt], false, b0,  (short)0, acc2[mt][0], false, false);
            acc2[mt][1] = __builtin_amdgcn_wmma_f32_16x16x4_f32(
                false, a[mt], false, b1f, (short)0, acc2[mt][1], false, false);
        }
    }
    {
        const float bb0 = bias2[nt0 * 16 + l16];
        const float bb1 = bias2[nt1 * 16 + l16];
#pragma unroll
        for (int mt = 0; mt < 4; ++mt) {
#pragma unroll
            for (int r = 0; r < 8; ++r) {
                int rr = mt * 16 + r + 8 * half;
                float o0 = acc2[mt][0][r] + bb0;
                float o1 = acc2[mt][1][r] + bb1;
                OUT[(size_t)(row0 + rr) * DD + nt0 * 16 + l16] = o0;
                OUT[(size_t)(row0 + rr) * DD + nt1 * 16 + l16] = o1;
                Xs[rr * LDSS + nt0 * 16 + l16] = o0;   // keep tile for norm pass
                Xs[rr * LDSS + nt1 * 16 + l16] = o1;
            }
        }
    }
    __syncthreads();

    // Row norms: 4 threads per row, each sums 64 elements.
    {
        int r = tid >> 2, q = tid & 3;
        float s = 0.0f;
        for (int c = q * 64; c < q * 64 + 64; ++c) {
            float v = Xs[r * LDSS + c];
            s += v * v;
        }
        Hs[tid] = s;
    }
    __syncthreads();
    if (tid < 64) {
        float s = Hs[tid * 4] + Hs[tid * 4 + 1] + Hs[tid * 4 + 2] + Hs[tid * 4 + 3];
        norms[row0 + tid] = sqrtf(s);
    }
}

// ---------------------------------------------------------------------------
// Kernel 2: per-graph [start,end) ranges by binary search on sorted batch;
// thread 0 also zeroes the output scalar (fresh every launch -> deterministic).
// ---------------------------------------------------------------------------
__global__ __launch_bounds__(512) void seg_kernel(
    const int* __restrict__ batch, int* __restrict__ segs, float* __restrict__ out)
{
    int g = threadIdx.x;
    if (g == 0) out[0] = 0.0f;
    if (g >= NG) return;
    int lo = 0, hi = NN;
    while (lo < hi) { int mid = (lo + hi) >> 1; if (batch[mid] < g) lo = mid + 1; else hi = mid; }
    int s = lo;
    hi = NN;
    while (lo < hi) { int mid = (lo + hi) >> 1; if (batch[mid] < g + 1) lo = mid + 1; else hi = mid; }
    segs[2 * g] = s;
    segs[2 * g + 1] = lo;
}

// ---------------------------------------------------------------------------
// Kernel 3: one workgroup (4 waves) per graph segment. Each wave computes
// 16x16 WMMA tiles of XP_seg @ YP_seg^T (K=256), scales by 1/(|x||y|+eps),
// exp(), LDS-atomic column sums + diagonal, then block-reduced
// -log(pos/(denom+eps))/N added to the global loss.
// ---------------------------------------------------------------------------
__global__ __launch_bounds__(128) void nce_kernel(
    const float* __restrict__ XP, const float* __restrict__ YP,
    const float* __restrict__ xn, const float* __restrict__ yn,
    const int* __restrict__ segs, float* __restrict__ out)
{
    __shared__ float denomS[128];
    __shared__ float posS[128];
    __shared__ float redS[128];

    const int g   = blockIdx.x;
    const int s   = segs[2 * g];
    const int e   = segs[2 * g + 1];
    const int n   = e - s;
    const int tid = threadIdx.x;

    denomS[tid] = 0.0f;
    posS[tid]   = 0.0f;
    __syncthreads();

    if (n > 0) {
        const int wave = tid >> 5;
        const int lane = tid & 31;
        const int half = lane >> 4;
        const int l16  = lane & 15;
        const int nmax = n > 128 ? 128 : n;   // capacity; Binomial(8192,1/512) never nears 128
        const int nt   = (nmax + 15) >> 4;
        const int ntiles = nt * nt;

        for (int tile = wave; tile < ntiles; tile += 4) {   // wave-uniform loop
            const int tm = tile / nt, tn = tile % nt;
            int rloc = tm * 16 + l16; rloc = rloc < nmax ? rloc : nmax - 1;  // clamp-pad
            int cloc = tn * 16 + l16; const int cc = cloc < nmax ? cloc : nmax - 1;
            const float* arow = XP + (size_t)(s + rloc) * DD;
            const float* brow = YP + (size_t)(s + cc) * DD;    // B = YP^T -> contiguous
            v8f acc = {};
#pragma unroll 4
            for (int kk = 0; kk < 64; ++kk) {
                const int kl = kk * 4 + 2 * half;
                v2f a = *(const v2f*)(arow + kl);
                v2f b = *(const v2f*)(brow + kl);
                acc = __builtin_amdgcn_wmma_f32_16x16x4_f32(
                    false, a, false, b, (short)0, acc, false, false);
            }
            const float ynj = yn[s + cc];
#pragma unroll
            for (int r = 0; r < 8; ++r) {
                const int rl = tm * 16 + r + 8 * half;
                if (rl < nmax && cloc < nmax) {
                    float sim = acc[r] / (xn[s + rl] * ynj + EPSV);
                    float ev  = expf(sim);           // TAU == 1
                    atomicAdd(&denomS[cloc], ev);    // ds_add_f32
                    if (rl == cloc) posS[cloc] = ev; // unique writer per diagonal
                }
            }
        }
    }
    __syncthreads();

    float contrib = 0.0f;
    if (tid < n && tid < 128)
        contrib = -logf(posS[tid] / (denomS[tid] + EPSV)) * (1.0f / (float)NN);
    redS[tid] = contrib;
    __syncthreads();
    if (tid == 0) {
        float ssum = 0.0f;
        for (int i = 0; i < 128; ++i) ssum += redS[i];
        unsafeAtomicAdd(out, ssum);                  // global_atomic_add_f32
    }
}

// ---------------------------------------------------------------------------
extern "C" void kernel_launch(void* const* d_in, const int* in_sizes, int n_in,
                              void* d_out, int out_size, void* d_ws, size_t ws_size,
                              hipStream_t stream) {
    const float* x   = (const float*)d_in[0];
    const float* y   = (const float*)d_in[1];
    const int*   bat = (const int*)  d_in[2];
    const float* Wx1 = (const float*)d_in[3];
    const float* bx1 = (const float*)d_in[4];
    const float* Wx2 = (const float*)d_in[5];
    const float* bx2 = (const float*)d_in[6];
    const float* Wy1 = (const float*)d_in[7];
    const float* by1 = (const float*)d_in[8];
    const float* Wy2 = (const float*)d_in[9];
    const float* by2 = (const float*)d_in[10];
    float* out = (float*)d_out;

    float* XP = (float*)d_ws;                 // 8192*256
    float* YP = XP + (size_t)NN * DD;         // 8192*256
    float* xn = YP + (size_t)NN * DD;         // 8192
    float* yn = xn + NN;                      // 8192
    int*   sg = (int*)(yn + NN);              // 2*512

    mlp_norm_kernel<<<NN / 64, 256, 0, stream>>>(x, Wx1, bx1, Wx2, bx2, XP, xn);
    mlp_norm_kernel<<<NN / 64, 256, 0, stream>>>(y, Wy1, by1, Wy2, by2, YP, yn);
    seg_kernel<<<1, 512, 0, stream>>>(bat, sg, out);
    nce_kernel<<<NG, 128, 0, stream>>>(XP, YP, xn, yn, sg, out);
}